// ModelEstimator_78305843741261
// MI455X (gfx1250) — compile-verified
//
#include <hip/hip_runtime.h>

typedef float v2f __attribute__((ext_vector_type(2)));
typedef float v8f __attribute__((ext_vector_type(8)));

#define N_PTS    1024
#define NTHREADS 256
#define NWAVES   8          // 256 threads / wave32
#define SQRT2F   1.41421356237309515f

__global__ __launch_bounds__(NTHREADS)
void fundamental_kernel(const float* __restrict__ kp1,
                        const float* __restrict__ kp2,
                        const float* __restrict__ wts,
                        float* __restrict__ out)
{
    __shared__ float s_k1[N_PTS * 3];      // 12 KB  (later overwritten with w*nk1x, w*nk1y, w)
    __shared__ float s_k2[N_PTS * 3];      // 12 KB  (later overwritten with nk2x, nk2y, z2)
    __shared__ float s_w [N_PTS];          // 4 KB
    __shared__ float s_tmp[5 * NTHREADS];  // 5 KB reduction scratch
    __shared__ float s_Mw[NWAVES * 256];   // 8 KB per-wave 16x16 Gram tiles
    __shared__ float s_M [81];             // 9x9 Gram matrix

    const int b = blockIdx.x;
    const int t = threadIdx.x;

    // ---------- Phase 0: stage batch into LDS (float4 vector loads) ----------
    {
        const float4* g1 = (const float4*)(kp1 + (size_t)b * N_PTS * 3);
        const float4* g2 = (const float4*)(kp2 + (size_t)b * N_PTS * 3);
        const float4* gw = (const float4*)(wts + (size_t)b * N_PTS);
        float4* l1 = (float4*)s_k1;
        float4* l2 = (float4*)s_k2;
        float4* lw = (float4*)s_w;
        for (int i = t; i < N_PTS * 3 / 4; i += NTHREADS) l1[i] = g1[i];
        for (int i = t; i < N_PTS * 3 / 4; i += NTHREADS) l2[i] = g2[i];
        for (int i = t; i < N_PTS / 4;     i += NTHREADS) lw[i] = gw[i];
    }
    __syncthreads();

    // ---------- Phase 1: weighted centers (5-way tree reduction) ----------
    {
        float p0 = 0.f, p1 = 0.f, p2 = 0.f, p3 = 0.f, p4 = 0.f;
        for (int p = t; p < N_PTS; p += NTHREADS) {
            float wv = s_w[p];
            p0 += wv;
            p1 += wv * s_k1[3 * p + 0];
            p2 += wv * s_k1[3 * p + 1];
            p3 += wv * s_k2[3 * p + 0];
            p4 += wv * s_k2[3 * p + 1];
        }
        s_tmp[0 * NTHREADS + t] = p0;
        s_tmp[1 * NTHREADS + t] = p1;
        s_tmp[2 * NTHREADS + t] = p2;
        s_tmp[3 * NTHREADS + t] = p3;
        s_tmp[4 * NTHREADS + t] = p4;
    }
    __syncthreads();
    for (int str = NTHREADS / 2; str > 0; str >>= 1) {
        if (t < str) {
            #pragma unroll
            for (int q = 0; q < 5; ++q)
                s_tmp[q * NTHREADS + t] += s_tmp[q * NTHREADS + t + str];
        }
        __syncthreads();
    }
    const float sumw = s_tmp[0 * NTHREADS];
    const float c1x  = s_tmp[1 * NTHREADS] / sumw;
    const float c1y  = s_tmp[2 * NTHREADS] / sumw;
    const float c2x  = s_tmp[3 * NTHREADS] / sumw;
    const float c2y  = s_tmp[4 * NTHREADS] / sumw;
    __syncthreads();

    // ---------- Phase 2: weighted mean distances -> scales ----------
    {
        float d1 = 0.f, d2 = 0.f;
        for (int p = t; p < N_PTS; p += NTHREADS) {
            float wv = s_w[p];
            float ax = s_k1[3 * p + 0] - c1x, ay = s_k1[3 * p + 1] - c1y;
            float bx = s_k2[3 * p + 0] - c2x, by = s_k2[3 * p + 1] - c2y;
            d1 += wv * sqrtf(ax * ax + ay * ay);
            d2 += wv * sqrtf(bx * bx + by * by);
        }
        s_tmp[0 * NTHREADS + t] = d1;
        s_tmp[1 * NTHREADS + t] = d2;
    }
    __syncthreads();
    for (int str = NTHREADS / 2; str > 0; str >>= 1) {
        if (t < str) {
            s_tmp[0 * NTHREADS + t] += s_tmp[0 * NTHREADS + t + str];
            s_tmp[1 * NTHREADS + t] += s_tmp[1 * NTHREADS + t + str];
        }
        __syncthreads();
    }
    const float sc1 = SQRT2F * sumw / s_tmp[0 * NTHREADS];
    const float sc2 = SQRT2F * sumw / s_tmp[1 * NTHREADS];
    __syncthreads();

    // ---------- Phase 3: in-place transform: left factors (w*nk1x, w*nk1y, w), right (nk2x, nk2y, z2)
    for (int p = t; p < N_PTS; p += NTHREADS) {
        float x1 = s_k1[3 * p + 0], y1 = s_k1[3 * p + 1], z1 = s_k1[3 * p + 2];
        float x2 = s_k2[3 * p + 0], y2 = s_k2[3 * p + 1], z2 = s_k2[3 * p + 2];
        float wv = s_w[p];
        s_k1[3 * p + 0] = wv * sc1 * (x1 - c1x * z1);
        s_k1[3 * p + 1] = wv * sc1 * (y1 - c1y * z1);
        s_k1[3 * p + 2] = wv;                       // homogeneous row block: w * 1
        s_k2[3 * p + 0] = sc2 * (x2 - c2x * z2);
        s_k2[3 * p + 1] = sc2 * (y2 - c2y * z2);
        s_k2[3 * p + 2] = z2;
    }
    __syncthreads();

    // ---------- Phase 4: M = A^T A via V_WMMA_F32_16X16X4_F32 symmetric rank-4 updates ----------
    // A operand (16x4): lane holds row m=l%16, K = vgpr + 2*(l/16)
    // B operand (4x16): lane holds col n=l%16, K = vgpr + 2*(l/16)  -> identical register image
    {
        const int wv   = t >> 5;
        const int lane = t & 31;
        const int half = lane >> 4;
        const int comp = lane & 15;
        const int ri   = comp / 3;                 // 0..5 (>=3 means zero pad row)
        const int ci   = comp % 3;
        const int riC  = (ri < 3) ? ri : 2;        // clamped LDS index
        const float lmask = (comp < 9) ? 1.f : 0.f;

        v8f acc = {};
        #pragma unroll 4
        for (int it = 0; it < 32; ++it) {
            int p0 = wv * 128 + it * 4 + 2 * half;  // this half-wave's first point
            float L0 = s_k1[3 * p0 + riC]     * lmask;
            float R0 = s_k2[3 * p0 + ci];
            float L1 = s_k1[3 * (p0+1) + riC] * lmask;
            float R1 = s_k2[3 * (p0+1) + ci];
            v2f a;
            a.x = L0 * R0;
            a.y = L1 * R1;
            // D(16x16) += A(16x4) x B(4x16), A and B share the same registers
            acc = __builtin_amdgcn_wmma_f32_16x16x4_f32(
                false, a, false, a, (short)0, acc, false, false);
        }
        // C/D layout: vgpr v, lanes 0-15 -> (m=v, n=lane); lanes 16-31 -> (m=v+8, n=lane-16)
        #pragma unroll
        for (int v = 0; v < 8; ++v) {
            int m = v + 8 * half;
            s_Mw[wv * 256 + m * 16 + comp] = acc[v];
        }
    }
    __syncthreads();

    if (t < 81) {
        int m = t / 9, n = t % 9;
        float s = 0.f;
        #pragma unroll
        for (int w8 = 0; w8 < NWAVES; ++w8)
            s += s_Mw[w8 * 256 + m * 16 + n];
        s_M[t] = s;
    }
    __syncthreads();

    // ---------- Phase 5: null vector (9x9 Jacobi), rank-2 projection, denormalize ----------
    if (t == 0) {
        float A[9][9], V[9][9];
        for (int i = 0; i < 9; ++i)
            for (int j = 0; j < 9; ++j) {
                A[i][j] = s_M[i * 9 + j];
                V[i][j] = (i == j) ? 1.f : 0.f;
            }
        for (int sweep = 0; sweep < 12; ++sweep) {
            for (int p = 0; p < 8; ++p)
                for (int q = p + 1; q < 9; ++q) {
                    float apq = A[p][q];
                    if (fabsf(apq) <= 1e-9f * (fabsf(A[p][p]) + fabsf(A[q][q]))) continue;
                    float tau = (A[q][q] - A[p][p]) / (2.f * apq);
                    float tt  = ((tau >= 0.f) ? 1.f : -1.f) /
                                (fabsf(tau) + sqrtf(1.f + tau * tau));
                    float c = 1.f / sqrtf(1.f + tt * tt);
                    float s = tt * c;
                    for (int k = 0; k < 9; ++k) {
                        float akp = A[k][p], akq = A[k][q];
                        A[k][p] = c * akp - s * akq;
                        A[k][q] = s * akp + c * akq;
                    }
                    for (int k = 0; k < 9; ++k) {
                        float apk = A[p][k], aqk = A[q][k];
                        A[p][k] = c * apk - s * aqk;
                        A[q][k] = s * apk + c * aqk;
                    }
                    for (int k = 0; k < 9; ++k) {
                        float vkp = V[k][p], vkq = V[k][q];
                        V[k][p] = c * vkp - s * vkq;
                        V[k][q] = s * vkp + c * vkq;
                    }
                }
        }
        int mi = 0;
        for (int i = 1; i < 9; ++i) if (A[i][i] < A[mi][mi]) mi = i;

        float F[3][3];
        for (int i = 0; i < 3; ++i)
            for (int j = 0; j < 3; ++j)
                F[i][j] = V[3 * i + j][mi];

        // G = F^T F, smallest eigenvector via 3x3 Jacobi
        float G[3][3], V3[3][3];
        for (int i = 0; i < 3; ++i)
            for (int j = 0; j < 3; ++j) {
                float s = 0.f;
                for (int k = 0; k < 3; ++k) s += F[k][i] * F[k][j];
                G[i][j] = s;
                V3[i][j] = (i == j) ? 1.f : 0.f;
            }
        for (int sweep = 0; sweep < 8; ++sweep) {
            for (int p = 0; p < 2; ++p)
                for (int q = p + 1; q < 3; ++q) {
                    float apq = G[p][q];
                    if (fabsf(apq) <= 1e-12f * (fabsf(G[p][p]) + fabsf(G[q][q]))) continue;
                    float tau = (G[q][q] - G[p][p]) / (2.f * apq);
                    float tt  = ((tau >= 0.f) ? 1.f : -1.f) /
                                (fabsf(tau) + sqrtf(1.f + tau * tau));
                    float c = 1.f / sqrtf(1.f + tt * tt);
                    float s = tt * c;
                    for (int k = 0; k < 3; ++k) {
                        float akp = G[k][p], akq = G[k][q];
                        G[k][p] = c * akp - s * akq;
                        G[k][q] = s * akp + c * akq;
                    }
                    for (int k = 0; k < 3; ++k) {
                        float apk = G[p][k], aqk = G[q][k];
                        G[p][k] = c * apk - s * aqk;
                        G[q][k] = s * apk + c * aqk;
                    }
                    for (int k = 0; k < 3; ++k) {
                        float vkp = V3[k][p], vkq = V3[k][q];
                        V3[k][p] = c * vkp - s * vkq;
                        V3[k][q] = s * vkp + c * vkq;
                    }
                }
        }
        int m3 = 0;
        for (int i = 1; i < 3; ++i) if (G[i][i] < G[m3][m3]) m3 = i;
        float v3[3] = { V3[0][m3], V3[1][m3], V3[2][m3] };

        // F_proj = F - (F v3) v3^T   (zeroes the smallest singular value)
        float Fv[3];
        for (int i = 0; i < 3; ++i)
            Fv[i] = F[i][0] * v3[0] + F[i][1] * v3[1] + F[i][2] * v3[2];
        float Fp[3][3];
        for (int i = 0; i < 3; ++i)
            for (int j = 0; j < 3; ++j)
                Fp[i][j] = F[i][j] - Fv[i] * v3[j];

        // out = T1^T Fp T2,  T = [[s,0,-s*cx],[0,s,-s*cy],[0,0,1]]
        float a1 = -sc1 * c1x, b1 = -sc1 * c1y;
        float a2 = -sc2 * c2x, b2 = -sc2 * c2y;
        float R[3][3];
        for (int j = 0; j < 3; ++j) {
            R[0][j] = sc1 * Fp[0][j];
            R[1][j] = sc1 * Fp[1][j];
            R[2][j] = a1 * Fp[0][j] + b1 * Fp[1][j] + Fp[2][j];
        }
        float* o = out + (size_t)b * 9;
        for (int i = 0; i < 3; ++i) {
            o[i * 3 + 0] = R[i][0] * sc2;
            o[i * 3 + 1] = R[i][1] * sc2;
            o[i * 3 + 2] = R[i][0] * a2 + R[i][1] * b2 + R[i][2];
        }
    }
}

extern "C" void kernel_launch(void* const* d_in, const int* in_sizes, int n_in,
                              void* d_out, int out_size, void* d_ws, size_t ws_size,
                              hipStream_t stream) {
    const float* kp1 = (const float*)d_in[0];
    const float* kp2 = (const float*)d_in[1];
    const float* wts = (const float*)d_in[2];
    float* out = (float*)d_out;
    const int B = in_sizes[2] / N_PTS;   // weights is (B, N)
    fundamental_kernel<<<B, NTHREADS, 0, stream>>>(kp1, kp2, wts, out);
}